// GraphEva_64828236366237
// MI455X (gfx1250) — compile-verified
//
#include <hip/hip_runtime.h>
#include <hip/hip_bf16.h>
#include <math.h>

typedef __attribute__((ext_vector_type(2))) float v2f;
typedef __attribute__((ext_vector_type(8))) float v8f;

#define KNOW   128
#define STU_N  50000
#define N_E    1000000
#define N_EP   950000
#define BATCH  8192
#define SOFT   0.2f
#define EPSF   1e-8f

// ---- workspace layout (in floats) ----
static const size_t OFF_AGGE = 0;
static const size_t OFF_AGGP = OFF_AGGE + (size_t)STU_N * KNOW;   // 6.4M
static const size_t OFF_DEGE = OFF_AGGP + (size_t)STU_N * KNOW;   // 12.8M
static const size_t OFF_DEGP = OFF_DEGE + STU_N;
static const size_t OFF_WC   = OFF_DEGP + STU_N;
static const size_t OFF_B2   = OFF_WC + (size_t)KNOW * KNOW;
static const size_t OFF_S    = OFF_B2 + (size_t)BATCH * KNOW;
static const size_t OFF_DIAG = OFF_S + KNOW;
static const size_t OFF_LOSS = OFF_DIAG + BATCH;
static const size_t WS_FLOATS = OFF_LOSS + 1;                     // ~14M floats (~56MB)

__device__ __forceinline__ void gAtomicAdd(float* p, float v) {
    unsafeAtomicAdd(p, v);   // native global_atomic_add_f32
}

__device__ __forceinline__ float waveAllSum(float v) {
    #pragma unroll
    for (int m = 1; m < 32; m <<= 1) v += __shfl_xor(v, m, 32);
    return v;
}

// ---------------- zero workspace ----------------
__global__ void zero_ws_k(float* ws, size_t n) {
    size_t i = (size_t)blockIdx.x * blockDim.x + threadIdx.x;
    size_t stride = (size_t)gridDim.x * blockDim.x;
    for (; i < n; i += stride) ws[i] = 0.0f;
}

// ---------------- Wc = W_ue + W_deep ----------------
__global__ void addw_k(const float* __restrict__ a, const float* __restrict__ b,
                       float* __restrict__ c) {
    int i = blockIdx.x * blockDim.x + threadIdx.x;
    if (i < KNOW * KNOW) c[i] = a[i] + b[i];
}

// ---------------- edge scatter: agg[dst] += exer[src]; deg[dst] += 1 ----------------
// one wave per edge; 32 lanes x float4 = 128 floats
__global__ void scatter_k(const float* __restrict__ exer, const int* __restrict__ src,
                          const int* __restrict__ dst, float* __restrict__ agg,
                          float* __restrict__ deg, int nEdges) {
    int e = (int)((blockIdx.x * (size_t)blockDim.x + threadIdx.x) >> 5);
    int lane = threadIdx.x & 31;
    if (e >= nEdges) return;
    int s = src[e];
    int d = dst[e];
    const float4 v = *(const float4*)(exer + (size_t)s * KNOW + lane * 4);
    float* base = agg + (size_t)d * KNOW + lane * 4;
    gAtomicAdd(base + 0, v.x);
    gAtomicAdd(base + 1, v.y);
    gAtomicAdd(base + 2, v.z);
    gAtomicAdd(base + 3, v.w);
    if (lane == 0) gAtomicAdd(deg + d, 1.0f);
}

// ---------------- stu2 = stu_emb + (aggE/deg) @ Wc ----------------
// grid.x = 3125 (M tiles of 16), block = 256 (8 waves -> 8 N tiles of 16)
__global__ void gemm_stu2_k(const float* __restrict__ aggE, const float* __restrict__ degE,
                            const float* __restrict__ Wc, const float* __restrict__ stu_emb,
                            float* __restrict__ out) {
    const int wave  = threadIdx.x >> 5;
    const int lane  = threadIdx.x & 31;
    const int hi    = lane >> 4;     // 0: K{0,1}, 1: K{2,3}
    const int l15   = lane & 15;
    const int mBase = blockIdx.x * 16;
    const int nBase = wave * 16;
    const int arow  = mBase + l15;
    const float invd = 1.0f / fmaxf(degE[arow], 1.0f);
    const float* aRow = aggE + (size_t)arow * KNOW;

    v8f c = {};
    for (int k0 = 0; k0 < KNOW; k0 += 4) {
        const int ka = k0 + hi * 2;
        v2f a, b;
        a.x = aRow[ka]     * invd;
        a.y = aRow[ka + 1] * invd;
        b.x = Wc[(size_t)ka       * KNOW + nBase + l15];
        b.y = Wc[(size_t)(ka + 1) * KNOW + nBase + l15];
        c = __builtin_amdgcn_wmma_f32_16x16x4_f32(false, a, false, b, (short)0, c, false, false);
    }
    #pragma unroll
    for (int j = 0; j < 8; ++j) {
        const int m = mBase + j + hi * 8;
        const size_t idx = (size_t)m * KNOW + nBase + l15;
        out[idx] = c[j] + stu_emb[idx];
    }
}

// ---------------- b2[i] = stu_emb[sid] + (aggP/degP)[sid]@W_ue_per + (aggE/degE)[sid]@W_deep ----
// grid.x = 512 (8192/16), block = 256
__global__ void gemm_b2_k(const float* __restrict__ aggE, const float* __restrict__ degE,
                          const float* __restrict__ aggP, const float* __restrict__ degP,
                          const float* __restrict__ Wper, const float* __restrict__ Wdeep,
                          const float* __restrict__ stu_emb, const int* __restrict__ stu_id,
                          float* __restrict__ b2) {
    const int wave  = threadIdx.x >> 5;
    const int lane  = threadIdx.x & 31;
    const int hi    = lane >> 4;
    const int l15   = lane & 15;
    const int mBase = blockIdx.x * 16;
    const int nBase = wave * 16;
    const int brow  = mBase + l15;          // batch row for this lane's A fragment
    const int sid   = stu_id[brow];
    const float invE = 1.0f / fmaxf(degE[sid], 1.0f);
    const float invP = 1.0f / fmaxf(degP[sid], 1.0f);
    const float* eRow = aggE + (size_t)sid * KNOW;
    const float* pRow = aggP + (size_t)sid * KNOW;

    v8f c = {};
    for (int k0 = 0; k0 < KNOW; k0 += 4) {
        const int ka = k0 + hi * 2;
        v2f ap, bp, ae, be;
        ap.x = pRow[ka] * invP;  ap.y = pRow[ka + 1] * invP;
        bp.x = Wper[(size_t)ka * KNOW + nBase + l15];
        bp.y = Wper[(size_t)(ka + 1) * KNOW + nBase + l15];
        c = __builtin_amdgcn_wmma_f32_16x16x4_f32(false, ap, false, bp, (short)0, c, false, false);
        ae.x = eRow[ka] * invE;  ae.y = eRow[ka + 1] * invE;
        be.x = Wdeep[(size_t)ka * KNOW + nBase + l15];
        be.y = Wdeep[(size_t)(ka + 1) * KNOW + nBase + l15];
        c = __builtin_amdgcn_wmma_f32_16x16x4_f32(false, ae, false, be, (short)0, c, false, false);
    }
    #pragma unroll
    for (int j = 0; j < 8; ++j) {
        const int outRow = mBase + j + hi * 8;
        const int osid = stu_id[outRow];
        const int n = nBase + l15;
        b2[(size_t)outRow * KNOW + n] = c[j] + stu_emb[(size_t)osid * KNOW + n];
    }
}

// ---------------- loss pass 1: per-row diag term + global sum of normalized b2 rows ----------
// one wave per row; 8 rows/block; grid.x = 1024
__global__ void loss1_k(const float* __restrict__ stu2, const float* __restrict__ b2,
                        const int* __restrict__ stu_id, float* __restrict__ sVec,
                        float* __restrict__ diag) {
    __shared__ float sblk[KNOW];
    const int tid = threadIdx.x;
    if (tid < KNOW) sblk[tid] = 0.0f;
    __syncthreads();

    const int row  = blockIdx.x * 8 + (tid >> 5);
    const int lane = tid & 31;
    const int sid  = stu_id[row];
    const float4 x1 = *(const float4*)(stu2 + (size_t)sid * KNOW + lane * 4);
    const float4 x2 = *(const float4*)(b2  + (size_t)row * KNOW + lane * 4);

    float n1 = x1.x*x1.x + x1.y*x1.y + x1.z*x1.z + x1.w*x1.w;
    float n2 = x2.x*x2.x + x2.y*x2.y + x2.z*x2.z + x2.w*x2.w;
    float dd = x1.x*x2.x + x1.y*x2.y + x1.z*x2.z + x1.w*x2.w;
    n1 = waveAllSum(n1);
    n2 = waveAllSum(n2);
    dd = waveAllSum(dd);
    const float inv1 = 1.0f / sqrtf(n1);
    const float inv2 = 1.0f / sqrtf(n2);
    if (lane == 0) diag[row] = dd * inv1 * inv2 / SOFT;

    atomicAdd(&sblk[lane * 4 + 0], x2.x * inv2);
    atomicAdd(&sblk[lane * 4 + 1], x2.y * inv2);
    atomicAdd(&sblk[lane * 4 + 2], x2.z * inv2);
    atomicAdd(&sblk[lane * 4 + 3], x2.w * inv2);
    __syncthreads();
    if (tid < KNOW) gAtomicAdd(&sVec[tid], sblk[tid]);
}

// ---------------- loss pass 2: row_sum via dot(n1, s); accumulate -log terms --------------
__global__ void loss2_k(const float* __restrict__ stu2, const float* __restrict__ sVec,
                        const float* __restrict__ diag, const int* __restrict__ stu_id,
                        float* __restrict__ lossAcc) {
    const int tid  = threadIdx.x;
    const int row  = blockIdx.x * 8 + (tid >> 5);
    const int lane = tid & 31;
    const int sid  = stu_id[row];
    const float4 x1 = *(const float4*)(stu2 + (size_t)sid * KNOW + lane * 4);
    const float4 sv = *(const float4*)(sVec + lane * 4);

    float n1 = x1.x*x1.x + x1.y*x1.y + x1.z*x1.z + x1.w*x1.w;
    float ds = x1.x*sv.x + x1.y*sv.y + x1.z*sv.z + x1.w*sv.w;
    n1 = waveAllSum(n1);
    ds = waveAllSum(ds);
    const float inv1   = 1.0f / sqrtf(n1);
    const float rowsum = ds * inv1 / SOFT;
    const float ratio  = expf(diag[row]) / (rowsum + EPSF);
    const float term   = -logf(fmaxf(ratio, EPSF));
    if (lane == 0) gAtomicAdd(lossAcc, term);
}

__global__ void finalize_k(const float* __restrict__ lossAcc, float* __restrict__ out) {
    out[(size_t)STU_N * KNOW] = *lossAcc / (float)BATCH;
}

extern "C" void kernel_launch(void* const* d_in, const int* in_sizes, int n_in,
                              void* d_out, int out_size, void* d_ws, size_t ws_size,
                              hipStream_t stream) {
    const float* stu_emb  = (const float*)d_in[0];
    const float* exer_emb = (const float*)d_in[1];
    const float* W_ue     = (const float*)d_in[2];
    const float* W_ue_per = (const float*)d_in[3];
    const float* W_deep   = (const float*)d_in[4];
    const int*   src_e    = (const int*)d_in[5];
    const int*   dst_e    = (const int*)d_in[6];
    const int*   src_p    = (const int*)d_in[7];
    const int*   dst_p    = (const int*)d_in[8];
    const int*   stu_id   = (const int*)d_in[9];

    float* out  = (float*)d_out;
    float* ws   = (float*)d_ws;
    float* aggE = ws + OFF_AGGE;
    float* aggP = ws + OFF_AGGP;
    float* degE = ws + OFF_DEGE;
    float* degP = ws + OFF_DEGP;
    float* Wc   = ws + OFF_WC;
    float* b2   = ws + OFF_B2;
    float* sVec = ws + OFF_S;
    float* diag = ws + OFF_DIAG;
    float* loss = ws + OFF_LOSS;

    // 1) zero accumulators (agg/deg/s/loss; rest is overwritten before read)
    zero_ws_k<<<2048, 256, 0, stream>>>(ws, WS_FLOATS);
    // 2) Wc = W_ue + W_deep
    addw_k<<<(KNOW * KNOW + 255) / 256, 256, 0, stream>>>(W_ue, W_deep, Wc);
    // 3) edge scatter (linear reassociation of the per-edge matmul)
    {
        size_t thr = (size_t)N_E * 32;
        scatter_k<<<(unsigned)((thr + 255) / 256), 256, 0, stream>>>(
            exer_emb, src_e, dst_e, aggE, degE, N_E);
        thr = (size_t)N_EP * 32;
        scatter_k<<<(unsigned)((thr + 255) / 256), 256, 0, stream>>>(
            exer_emb, src_p, dst_p, aggP, degP, N_EP);
    }
    // 4) stu2 (first output) via f32 WMMA
    gemm_stu2_k<<<STU_N / 16, 256, 0, stream>>>(aggE, degE, Wc, stu_emb, out);
    // 5) gathered b2 rows via f32 WMMA (two accumulating WMMAs per K-step)
    gemm_b2_k<<<BATCH / 16, 256, 0, stream>>>(aggE, degE, aggP, degP,
                                              W_ue_per, W_deep, stu_emb, stu_id, b2);
    // 6) loss reductions
    loss1_k<<<BATCH / 8, 256, 0, stream>>>(out, b2, stu_id, sVec, diag);
    loss2_k<<<BATCH / 8, 256, 0, stream>>>(out, sVec, diag, stu_id, loss);
    finalize_k<<<1, 1, 0, stream>>>(loss, out);
}